// MCSoftContrastiveLoss_43018392437306
// MI455X (gfx1250) — compile-verified
//
#include <hip/hip_runtime.h>
#include <hip/hip_bf16.h>

typedef __attribute__((ext_vector_type(2))) float v2f;
typedef __attribute__((ext_vector_type(8))) float v8f;
typedef int v4i_gnu __attribute__((__vector_size__(16)));   // int4, matches builtin param

#define NROW 4096      // N*K = 512*8
#define DDIM 1024
#define LDA  36        // padded LDS row stride (floats): conflict-free + 16B-aligned rows

// ---- CDNA5 async global->LDS copy (ASYNCcnt-tracked, no VGPR round trip) ----
__device__ __forceinline__ void async_copy_b128(const float* gsrc, float* lds_dst) {
#if __has_builtin(__builtin_amdgcn_global_load_async_to_lds_b128)
    typedef __attribute__((address_space(1))) v4i_gnu GV4;
    typedef __attribute__((address_space(3))) v4i_gnu LV4;
    __builtin_amdgcn_global_load_async_to_lds_b128(
        (GV4*)(__UINTPTR_TYPE__)gsrc,
        (LV4*)(unsigned)(__UINTPTR_TYPE__)lds_dst, 0, 0);
#else
    const unsigned l = (unsigned)(__UINTPTR_TYPE__)lds_dst;  // LDS offset = addr[31:0]
    asm volatile("global_load_async_to_lds_b128 %0, %1, off"
                 :: "v"(l), "v"(gsrc) : "memory");
#endif
}

__device__ __forceinline__ void wait_async0() {
#if __has_builtin(__builtin_amdgcn_s_wait_asynccnt)
    __builtin_amdgcn_s_wait_asynccnt(0);
#else
    asm volatile("s_wait_asynccnt 0x0" ::: "memory");
#endif
}

__global__ void mcsc_init_out(float* out) {
    if (threadIdx.x == 0 && blockIdx.x == 0) out[0] = 0.0f;
}

// Build a = mu + eps*exp(logsigma) (and b), plus row sums of squares sa/sb.
// grid: (4096, 2) blocks of 256; blockIdx.y: 0 = img->a, 1 = cap->b
__global__ __launch_bounds__(256) void mcsc_build_ab(
    const float* __restrict__ img_mean, const float* __restrict__ img_ls,
    const float* __restrict__ cap_mean, const float* __restrict__ cap_ls,
    const float* __restrict__ eps_img,  const float* __restrict__ eps_cap,
    float* __restrict__ a, float* __restrict__ b,
    float* __restrict__ sa, float* __restrict__ sb)
{
    const int row   = blockIdx.x;          // 0..4095  (= n*8 + ksample)
    const int which = blockIdx.y;          // 0: img, 1: cap
    const float* mean = which ? cap_mean : img_mean;
    const float* ls   = which ? cap_ls   : img_ls;
    const float* eps  = which ? eps_cap  : eps_img;
    float* dst  = which ? b  : a;
    float* sdst = which ? sb : sa;

    const int n   = row >> 3;
    const int tid = threadIdx.x;

    float acc = 0.0f;
#pragma unroll
    for (int t = 0; t < 4; ++t) {
        const int d = tid + t * 256;
        const float mu = mean[n * DDIM + d];
        const float sg = expf(ls[n * DDIM + d]);
        const float v  = fmaf(eps[(size_t)row * DDIM + d], sg, mu);
        dst[(size_t)row * DDIM + d] = v;
        acc = fmaf(v, v, acc);
    }
#pragma unroll
    for (int o = 16; o > 0; o >>= 1) acc += __shfl_xor(acc, o, 32);
    __shared__ float red[8];
    const int wave = tid >> 5, lane = tid & 31;
    if (lane == 0) red[wave] = acc;
    __syncthreads();
    if (tid == 0) {
        float s = 0.0f;
#pragma unroll
        for (int i = 0; i < 8; ++i) s += red[i];
        sdst[row] = s;
    }
}

// Fused GEMM (fp32 WMMA, async-LDS double buffered) + contrastive-loss epilogue.
// grid: (64, 64) blocks of 256 threads (8 wave32s). Block tile: 64x64 of the
// 4096x4096 Gram matrix; wave w computes tiles (w&3, 2*(w>>2)+{0,1}) of 16x16.
__global__ __launch_bounds__(256) void mcsc_gemm_loss(
    const float* __restrict__ A, const float* __restrict__ B,
    const float* __restrict__ sa, const float* __restrict__ sb,
    const float* __restrict__ shiftp, const float* __restrict__ negsp,
    float* __restrict__ out)
{
    __shared__ float As[2][64 * LDA];
    __shared__ float Bs[2][64 * LDA];
    __shared__ float partial[64];

    const int tid  = threadIdx.x;
    const int wave = tid >> 5;
    const int lane = tid & 31;
    const int R0 = blockIdx.y * 64;
    const int C0 = blockIdx.x * 64;

    const int tr = wave & 3;          // row tile 0..3
    const int tc = (wave >> 2) * 2;   // col tile pair base 0 or 2

    const int lm = lane & 15;         // M (or N) index within 16x16 fragment
    const int lk = (lane >> 4) * 2;   // K base offset for this half-wave

    // Per-thread staging slots: 512 float4 per 64x32 panel, 2 per thread.
    const int s0 = tid,        r0 = s0 >> 3, c0 = (s0 & 7) * 4;
    const int s1 = tid + 256,  r1 = s1 >> 3, c1 = (s1 & 7) * 4;
    const float* gA0 = A + (size_t)(R0 + r0) * DDIM + c0;
    const float* gA1 = A + (size_t)(R0 + r1) * DDIM + c1;
    const float* gB0 = B + (size_t)(C0 + r0) * DDIM + c0;
    const float* gB1 = B + (size_t)(C0 + r1) * DDIM + c1;

    v8f acc0 = {};
    v8f acc1 = {};

    // Prologue: async-fill buffer 0 with the first 64x32 panels.
    async_copy_b128(gA0, &As[0][r0 * LDA + c0]);
    async_copy_b128(gA1, &As[0][r1 * LDA + c1]);
    async_copy_b128(gB0, &Bs[0][r0 * LDA + c0]);
    async_copy_b128(gB1, &Bs[0][r1 * LDA + c1]);

    for (int kb = 0; kb < DDIM; kb += 32) {
        const int cur = (kb >> 5) & 1;
        wait_async0();        // this wave's pending copies (current buffer) done
        __syncthreads();      // all waves' copies visible; prev compute finished

        if (kb + 32 < DDIM) { // issue next panel into the other buffer
            const int nxt = cur ^ 1;
            async_copy_b128(gA0 + kb + 32, &As[nxt][r0 * LDA + c0]);
            async_copy_b128(gA1 + kb + 32, &As[nxt][r1 * LDA + c1]);
            async_copy_b128(gB0 + kb + 32, &Bs[nxt][r0 * LDA + c0]);
            async_copy_b128(gB1 + kb + 32, &Bs[nxt][r1 * LDA + c1]);
        }

        const float* as = As[cur];
        const float* bs = Bs[cur];
#pragma unroll
        for (int kk = 0; kk < 32; kk += 4) {
            const v2f af  = *(const v2f*)(&as[(tr * 16 + lm) * LDA + kk + lk]);
            const v2f bf0 = *(const v2f*)(&bs[(tc * 16 + lm) * LDA + kk + lk]);
            const v2f bf1 = *(const v2f*)(&bs[((tc + 1) * 16 + lm) * LDA + kk + lk]);
            acc0 = __builtin_amdgcn_wmma_f32_16x16x4_f32(false, af, false, bf0,
                                                         (short)0, acc0, false, false);
            acc1 = __builtin_amdgcn_wmma_f32_16x16x4_f32(false, af, false, bf1,
                                                         (short)0, acc1, false, false);
        }
    }

    // ---- fused epilogue ----
    const float sh = shiftp[0];
    const float ns = negsp[0];
    const float log64 = 4.15888308335967186f;   // log(K*K) = log(64)

    // C/D layout: lane holds column (C0 + tile*16 + lm), rows rowBase..rowBase+7
    const int rowBase = R0 + tr * 16 + ((lane >> 4) << 3);
    const int i = rowBase >> 3;

#pragma unroll
    for (int t = 0; t < 2; ++t) {
        const v8f acc = t ? acc1 : acc0;
        const int col = C0 + (tc + t) * 16 + lm;
        const int j   = col >> 3;
        const float sbc = sb[col];
        const float sgn = (i == j) ? 1.0f : -1.0f;

        float vals[8];
        float mx = -INFINITY;
#pragma unroll
        for (int r = 0; r < 8; ++r) {
            float d2 = sa[rowBase + r] + sbc - 2.0f * acc[r];
            d2 = fmaxf(d2, 0.0f);
            const float s  = fmaf(-ns, sqrtf(d2), sh);          // logits
            const float as_ = fabsf(s);
            const float denom = as_ + log1pf(expf(-2.0f * as_)); // logaddexp(s,-s)
            const float v = sgn * s - denom;
            vals[r] = v;
            mx = fmaxf(mx, v);
        }
        // logsumexp over the 8x8 (i,j) block: 8 regs x 8 contiguous lanes
#pragma unroll
        for (int o = 1; o < 8; o <<= 1) mx = fmaxf(mx, __shfl_xor(mx, o, 32));
        float se = 0.0f;
#pragma unroll
        for (int r = 0; r < 8; ++r) se += expf(vals[r] - mx);
#pragma unroll
        for (int o = 1; o < 8; o <<= 1) se += __shfl_xor(se, o, 32);

        const float nll = log64 - (mx + logf(se));
        if ((lane & 7) == 0)
            partial[wave * 8 + t * 4 + (lane >> 3)] = 2.0f * nll;
    }

    __syncthreads();
    if (tid < 32) {
        float v = partial[tid] + partial[tid + 32];
#pragma unroll
        for (int o = 16; o > 0; o >>= 1) v += __shfl_xor(v, o, 32);
        if (tid == 0) atomicAdd(out, v);
    }
}

extern "C" void kernel_launch(void* const* d_in, const int* in_sizes, int n_in,
                              void* d_out, int out_size, void* d_ws, size_t ws_size,
                              hipStream_t stream) {
    (void)in_sizes; (void)n_in; (void)out_size; (void)ws_size;
    const float* img_mean = (const float*)d_in[0];
    const float* img_ls   = (const float*)d_in[1];
    const float* cap_mean = (const float*)d_in[2];
    const float* cap_ls   = (const float*)d_in[3];
    const float* eps_img  = (const float*)d_in[4];
    const float* eps_cap  = (const float*)d_in[5];
    const float* shift    = (const float*)d_in[6];
    const float* negscale = (const float*)d_in[7];
    float* out = (float*)d_out;

    float* ws = (float*)d_ws;
    float* a  = ws;                                   // 4096*1024
    float* b  = a  + (size_t)NROW * DDIM;             // 4096*1024
    float* sa = b  + (size_t)NROW * DDIM;             // 4096
    float* sb = sa + NROW;                            // 4096

    mcsc_init_out<<<1, 1, 0, stream>>>(out);
    mcsc_build_ab<<<dim3(NROW, 2), 256, 0, stream>>>(
        img_mean, img_ls, cap_mean, cap_ls, eps_img, eps_cap, a, b, sa, sb);
    mcsc_gemm_loss<<<dim3(64, 64), 256, 0, stream>>>(
        a, b, sa, sb, shift, negscale, out);
}